// GCN_6794638262379
// MI455X (gfx1250) — compile-verified
//
#include <hip/hip_runtime.h>
#include <hip/hip_bf16.h>

typedef float v2f __attribute__((ext_vector_type(2)));
typedef float v8f __attribute__((ext_vector_type(8)));

// ---------------------------------------------------------------- degrees ---
__global__ void init_deg_kernel(float* __restrict__ deg, int n) {
  int i = blockIdx.x * blockDim.x + threadIdx.x;
  if (i < n) deg[i] = 1.0f;  // self-loop contributes 1 to every node
}

__global__ void edge_deg_kernel(const int* __restrict__ dst, float* __restrict__ deg, int nE) {
  int e = blockIdx.x * blockDim.x + threadIdx.x;
  if (e < nE) atomicAdd(&deg[dst[e]], 1.0f);
}

__global__ void dinv_kernel(const float* __restrict__ deg, float* __restrict__ dinv, int n) {
  int i = blockIdx.x * blockDim.x + threadIdx.x;
  if (i < n) {
    float d = deg[i];
    dinv[i] = (d > 0.0f) ? rsqrtf(d) : 0.0f;
  }
}

// ------------------------------------------------------- dense GEMM (WMMA) ---
// H[N x NOUT] = X[N x 64] @ W[64 x NOUT], f32, V_WMMA_F32_16X16X4_F32.
// One wave per 16x16 output tile; (NOUT/16) waves per block cover 16 rows.
// A 16x4 f32 layout: lanes 0-15 -> M=lane, VGPR{0,1}=K{0,1}; lanes 16-31 -> K{2,3}.
// B 4x16 f32 layout: VGPR0 lanes0-15=K0, lanes16-31=K2; VGPR1 -> K1/K3.
// C 16x16 f32 layout: VGPR v -> M=v (lanes 0-15) / M=v+8 (lanes 16-31), N=lane&15.
template <int NOUT>
__global__ __launch_bounds__((NOUT / 16) * 32)
void gemm_wmma_kernel(const float* __restrict__ X, const float* __restrict__ W,
                      float* __restrict__ H, int nrows) {
  const int lane = threadIdx.x & 31;
  const int wave = threadIdx.x >> 5;
  const int hi = lane >> 4;   // 0: K pair {k,k+1}, 1: K pair {k+2,k+3}
  const int lo = lane & 15;
  const int rowBase = blockIdx.x * 16;
  const int colBase = wave * 16;

  int row = rowBase + lo;
  if (row >= nrows) row = nrows - 1;  // clamp (no branch) -> EXEC stays all-1s
  const int col = colBase + lo;
  const float* xr = X + (size_t)row * 64;

  v8f c = {0.f, 0.f, 0.f, 0.f, 0.f, 0.f, 0.f, 0.f};
#pragma unroll
  for (int k = 0; k < 64; k += 4) {
    const int ka = k + 2 * hi;
    v2f a, b;
    a.x = xr[ka];
    a.y = xr[ka + 1];
    b.x = W[(size_t)ka * NOUT + col];
    b.y = W[(size_t)(ka + 1) * NOUT + col];
    c = __builtin_amdgcn_wmma_f32_16x16x4_f32(false, a, false, b, (short)0, c,
                                              false, false);
  }
#pragma unroll
  for (int v = 0; v < 8; ++v) {
    int r = rowBase + hi * 8 + v;
    if (r < nrows) H[(size_t)r * NOUT + col] = c[v];
  }
}

// -------------------------------------------------------- edge aggregation ---
// agg[i,:] = dinv[i]^2 * h[i,:]   (self-loop term; also initializes agg)
template <int NOUT>
__global__ void self_loop_kernel(const float* __restrict__ dinv,
                                 const float* __restrict__ h,
                                 float* __restrict__ agg, long long total) {
  long long t = blockIdx.x * (long long)blockDim.x + threadIdx.x;
  if (t >= total) return;
  int i = (int)(t / NOUT);
  float di = dinv[i];
  agg[t] = di * di * h[t];
}

// agg[dst,:] += dinv[src]*dinv[dst] * h[src,:]  — one thread per (edge, col)
template <int NOUT>
__global__ void edge_scatter_kernel(const int* __restrict__ src, const int* __restrict__ dst,
                                    const float* __restrict__ dinv,
                                    const float* __restrict__ h,
                                    float* __restrict__ agg, int nE) {
  long long t = blockIdx.x * (long long)blockDim.x + threadIdx.x;
  int e = (int)(t / NOUT);
  int j = (int)(t % NOUT);
  if (e >= nE) return;
  int s = src[e];
  int d = dst[e];
  float w = dinv[s] * dinv[d];
  atomicAdd(agg + (size_t)d * NOUT + j, w * h[(size_t)s * NOUT + j]);
}

// out = (RELU ? max(agg + b, 0) : agg + b)
template <int NOUT, bool RELU>
__global__ void bias_act_kernel(const float* __restrict__ agg, const float* __restrict__ b,
                                float* __restrict__ out, long long total) {
  long long t = blockIdx.x * (long long)blockDim.x + threadIdx.x;
  if (t >= total) return;
  int j = (int)(t % NOUT);
  float v = agg[t] + b[j];
  out[t] = RELU ? fmaxf(v, 0.0f) : v;
}

// -------------------------------------------------------------------- host ---
extern "C" void kernel_launch(void* const* d_in, const int* in_sizes, int n_in,
                              void* d_out, int out_size, void* d_ws, size_t ws_size,
                              hipStream_t stream) {
  const float* x  = (const float*)d_in[0];
  const int*   ei = (const int*)d_in[1];   // int32: [src(E), dst(E)]
  const float* W1 = (const float*)d_in[2];
  const float* b1 = (const float*)d_in[3];
  const float* W2 = (const float*)d_in[4];
  const float* b2 = (const float*)d_in[5];
  const float* W3 = (const float*)d_in[6];
  const float* b3 = (const float*)d_in[7];

  const int N = in_sizes[0] / 64;
  const int E = in_sizes[1] / 2;
  const int* src = ei;
  const int* dst = ei + E;

  float* deg  = (float*)d_ws;              // N
  float* dinv = deg + N;                   // N
  float* bufA = dinv + N;                  // N*64
  float* bufB = bufA + (size_t)N * 64;     // N*64

  const int TB = 256;
  auto nb = [](long long t, int tb) { return (unsigned)((t + tb - 1) / tb); };
  const long long tot64 = (long long)N * 64;
  const long long tot32 = (long long)N * 32;
  const long long e64 = (long long)E * 64;
  const long long e32 = (long long)E * 32;
  const unsigned rowTiles = (unsigned)((N + 15) / 16);

  // symmetric-normalization coefficients
  init_deg_kernel<<<nb(N, TB), TB, 0, stream>>>(deg, N);
  edge_deg_kernel<<<nb(E, TB), TB, 0, stream>>>(dst, deg, E);
  dinv_kernel<<<nb(N, TB), TB, 0, stream>>>(deg, dinv, N);

  // Layer 1: h1 = x@W1 -> bufA; agg -> bufB; act1 -> bufA
  gemm_wmma_kernel<64><<<rowTiles, 128, 0, stream>>>(x, W1, bufA, N);
  self_loop_kernel<64><<<nb(tot64, TB), TB, 0, stream>>>(dinv, bufA, bufB, tot64);
  edge_scatter_kernel<64><<<nb(e64, TB), TB, 0, stream>>>(src, dst, dinv, bufA, bufB, E);
  bias_act_kernel<64, true><<<nb(tot64, TB), TB, 0, stream>>>(bufB, b1, bufA, tot64);

  // Layer 2: h2 = act1@W2 -> bufB; agg -> bufA; act2 -> bufB
  gemm_wmma_kernel<64><<<rowTiles, 128, 0, stream>>>(bufA, W2, bufB, N);
  self_loop_kernel<64><<<nb(tot64, TB), TB, 0, stream>>>(dinv, bufB, bufA, tot64);
  edge_scatter_kernel<64><<<nb(e64, TB), TB, 0, stream>>>(src, dst, dinv, bufB, bufA, E);
  bias_act_kernel<64, true><<<nb(tot64, TB), TB, 0, stream>>>(bufA, b2, bufB, tot64);

  // Layer 3: h3 = act2@W3 -> bufA (N x 32); agg -> bufB; out = agg + b3
  gemm_wmma_kernel<32><<<rowTiles, 64, 0, stream>>>(bufB, W3, bufA, N);
  self_loop_kernel<32><<<nb(tot32, TB), TB, 0, stream>>>(dinv, bufA, bufB, tot32);
  edge_scatter_kernel<32><<<nb(e32, TB), TB, 0, stream>>>(src, dst, dinv, bufA, bufB, E);
  bias_act_kernel<32, false><<<nb(tot32, TB), TB, 0, stream>>>(bufB, b3, (float*)d_out, tot32);
}